// QuerySpecificClusterModel_10531259809849
// MI455X (gfx1250) — compile-verified
//
#include <hip/hip_runtime.h>
#include <math.h>

// ---------------------------------------------------------------------------
// QuerySpecificClusterModel loss on MI455X (gfx1250)
//   B=256 samples, n=64 passages, d=768.
//   - TDM (tensor_load_to_lds) stages each sample's 64x128 K-slab in LDS once
//     (1x global traffic), padded to a 132-dword row stride (bank-conflict
//     free fragment gathers).
//   - Gram matrix per sample via V_WMMA_F32_16X16X4_F32; each wave owns two
//     16x16 tiles sharing the B fragment.
//   - Distances + single-linkage clustering in LDS, deterministic 2-kernel
//     reduction (no atomics -> graph-capture safe, bitwise repeatable).
// ---------------------------------------------------------------------------

typedef float v2f  __attribute__((ext_vector_type(2)));
typedef float v8f  __attribute__((ext_vector_type(8)));
typedef unsigned int u32x4 __attribute__((ext_vector_type(4)));
typedef int   i32x4 __attribute__((ext_vector_type(4)));
typedef int   i32x8 __attribute__((ext_vector_type(8)));

#define NB   256
#define NP   64
#define ND   768
#define KB   128            // K columns staged per TDM transfer
#define PADW 4              // TDM pad: 4 dwords after every 128 dwords
#define LROW (KB + PADW)    // 132-dword staged row stride
#define BIGV 1e9f

__global__ __launch_bounds__(256) void qscm_batch_kernel(
    const float* __restrict__ q,      // [NB, ND]
    const float* __restrict__ psg,    // [NB, NP, ND]
    const int*   __restrict__ labels, // [NB, NP]
    float*       __restrict__ ws)     // per-batch partials: sim, cnt, dis, err
{
    __shared__ __align__(16) float sh_s[NP * LROW]; // staged raw psg K-slab
    __shared__ __align__(16) float sh_q[ND];        // query row
    __shared__ float Dmat[NP * 65];                 // 64x64 dists, stride 65
    __shared__ float sq[NP];                        // row squared norms
    __shared__ int   tl[NP];                        // true labels
    __shared__ int   cl[NP];                        // cluster representatives
    __shared__ int   act[NP];                       // active flags
    __shared__ float red [256];
    __shared__ int   redi[256];
    __shared__ int   sh_k;                          // unique-label count

    const int b    = blockIdx.x;
    const int t    = threadIdx.x;
    const int wave = t >> 5;
    const int lane = t & 31;

    // ---- stage 0: q + labels to LDS ---------------------------------------
    for (int i = t; i < ND; i += 256) sh_q[i] = q[(size_t)b * ND + i];
    if (t < NP) tl[t] = labels[b * NP + t];
    if (t == 0) sh_k = 0;

    // Wave tile assignment: tile0 = wave = (ti0,tj), tile1 = wave+8 = (ti0+2,tj)
    const int ti0 = wave >> 2, tj = wave & 3;
    const int rA0 = ti0 * 16 + (lane & 15);
    const int rA1 = (ti0 + 2) * 16 + (lane & 15);
    const int rB  = tj * 16 + (lane & 15);
    const int ko  = (lane < 16) ? 0 : 2;       // K half per ISA fragment layout

    v8f acc0 = {0.f, 0.f, 0.f, 0.f, 0.f, 0.f, 0.f, 0.f};
    v8f acc1 = {0.f, 0.f, 0.f, 0.f, 0.f, 0.f, 0.f, 0.f};
    float nsum = 0.f;                          // row-norm partial
    const int nrow = t >> 2, npart = t & 3;

    const unsigned lds_base = (unsigned)(size_t)(void*)sh_s; // LDS byte offset

    // ---- stage 1: K-block loop: TDM stage -> norms + WMMA from LDS --------
    for (int kb = 0; kb < ND; kb += KB) {
        // Wave 0 issues the Tensor Data Mover transfer:
        //   2D tile: 64 rows x 128 cols, f32, tensor stride 768,
        //   LDS pad 4 dwords per 128 dwords -> 132-dword row stride.
        if (t < 32) {
            unsigned long long ga =
                (unsigned long long)(size_t)(psg + (size_t)b * NP * ND + kb);
            u32x4 g0;
            g0[0] = 1u;                                   // count=1, user mode
            g0[1] = lds_base;                             // lds_addr
            g0[2] = (unsigned)(ga & 0xFFFFFFFFu);         // global_addr[31:0]
            g0[3] = (unsigned)((ga >> 32) & 0x01FFFFFFu)  // global_addr[56:32]
                  | (2u << 30);                           // type=2 (image)
            i32x8 g1;
            g1[0] = (2 << 16)      // data_size = 4B
                  | (1 << 20)      // pad_enable
                  | (6 << 22)      // pad_interval: 128 dwords
                  | (3 << 25);     // pad_amount: 4 dwords
            g1[1] = (int)((ND & 0xFFFF) << 16);  // tensor_dim0 = 768 (lo16)
            g1[2] = (int)(NP << 16);             // dim0 hi=0 | tensor_dim1=64
            g1[3] = (int)(KB << 16);             // dim1 hi=0 | tile_dim0=128
            g1[4] = NP;                          // tile_dim1=64 | tile_dim2=0
            g1[5] = ND;                          // tensor_dim0_stride = 768
            g1[6] = 0;
            g1[7] = 0;
            i32x4 z4 = {0, 0, 0, 0};             // groups 2/3 unused (2D)
            i32x8 z8 = {0, 0, 0, 0, 0, 0, 0, 0}; // trailing group (6-arg form)
            __builtin_amdgcn_tensor_load_to_lds(g0, g1, z4, z4, z8, 0);
            __builtin_amdgcn_s_wait_tensorcnt(0);
        }
        __syncthreads();   // publish staged slab (also covers stage-0 stores)

        // Row-norm partial from the staged slab (4 threads per row).
        {
            const float* pr = sh_s + nrow * LROW;
            const int c0 = npart * 32;
#pragma unroll 8
            for (int c = c0; c < c0 + 32; ++c) {
                float v = sh_q[kb + c] * pr[c];
                nsum += v * v;
            }
        }

        // WMMA partial products for this wave's two tiles (shared B fragment).
#pragma unroll 4
        for (int k = 0; k < KB; k += 4) {
            v2f qv = *(const v2f*)(sh_q + kb + k + ko);
            v2f bR = *(const v2f*)(sh_s + rB  * LROW + k + ko);
            v2f a0 = *(const v2f*)(sh_s + rA0 * LROW + k + ko);
            v2f a1 = *(const v2f*)(sh_s + rA1 * LROW + k + ko);
            v2f bv  = bR * qv;
            v2f av0 = a0 * qv;
            v2f av1 = a1 * qv;
            acc0 = __builtin_amdgcn_wmma_f32_16x16x4_f32(
                false, av0, false, bv, (short)0, acc0, false, false);
            acc1 = __builtin_amdgcn_wmma_f32_16x16x4_f32(
                false, av1, false, bv, (short)0, acc1, false, false);
        }
        __syncthreads();   // readers done before next TDM overwrites the slab
    }

    // ---- stage 2: finish norms, unique-label count ------------------------
    red[t] = nsum;
    __syncthreads();
    if ((t & 3) == 0) sq[t >> 2] = red[t] + red[t + 1] + red[t + 2] + red[t + 3];
    if (t == 0) {
        unsigned m = 0;
        for (int i = 0; i < NP; ++i) m |= 1u << (tl[i] & 31);
        sh_k = __popc(m);
    }
    __syncthreads();

    // ---- stage 3: write distance matrix from WMMA accumulators ------------
    {
        const int mo = (lane < 16) ? 0 : 8;
        const int gj = tj * 16 + (lane & 15);
#pragma unroll
        for (int v = 0; v < 8; ++v) {
            int gi = ti0 * 16 + mo + v;
            float d2 = sq[gi] + sq[gj] - 2.f * acc0[v];
            Dmat[gi * 65 + gj] = (d2 > 0.f) ? sqrtf(d2) : 0.f;
            gi = (ti0 + 2) * 16 + mo + v;
            d2 = sq[gi] + sq[gj] - 2.f * acc1[v];
            Dmat[gi * 65 + gj] = (d2 > 0.f) ? sqrtf(d2) : 0.f;
        }
    }
    __syncthreads();

    // ---- stage 4: similar / dissimilar distance partial sums --------------
    float ssim = 0.f, sdis = 0.f, scnt = 0.f;
    for (int idx = t; idx < NP * NP; idx += 256) {
        const int i = idx >> 6, j = idx & 63;
        const float d = Dmat[i * 65 + j];
        if (tl[i] == tl[j]) { ssim += d; scnt += 1.f; }
        else                { sdis += d; }
    }
    red[t] = ssim; __syncthreads();
    for (int o = 128; o; o >>= 1) { if (t < o) red[t] += red[t + o]; __syncthreads(); }
    if (t == 0) ws[b] = red[0];
    __syncthreads();
    red[t] = scnt; __syncthreads();
    for (int o = 128; o; o >>= 1) { if (t < o) red[t] += red[t + o]; __syncthreads(); }
    if (t == 0) ws[NB + b] = red[0];
    __syncthreads();
    red[t] = sdis; __syncthreads();
    for (int o = 128; o; o >>= 1) { if (t < o) red[t] += red[t + o]; __syncthreads(); }
    if (t == 0) ws[2 * NB + b] = red[0];
    __syncthreads();

    // ---- stage 5: single-linkage agglomeration, stop at k clusters --------
    if (t < NP) { act[t] = 1; cl[t] = t; Dmat[t * 65 + t] = BIGV; }
    __syncthreads();

    const int nmerges = NP - sh_k;            // labels after (n-k) merges
    for (int step = 0; step < nmerges; ++step) {
        // block-wide argmin with first-flat-index tie-break (== jnp.argmin)
        float best = BIGV; int bidx = 1 << 30;
        for (int idx = t; idx < NP * NP; idx += 256) {
            const int i = idx >> 6, j = idx & 63;
            if (i != j && act[i] && act[j]) {
                const float v = Dmat[i * 65 + j];
                if (v < best || (v == best && idx < bidx)) { best = v; bidx = idx; }
            }
        }
        red[t] = best; redi[t] = bidx;
        __syncthreads();
        for (int o = 128; o; o >>= 1) {
            if (t < o) {
                const float v = red[t + o]; const int ix = redi[t + o];
                if (v < red[t] || (v == red[t] && ix < redi[t])) { red[t] = v; redi[t] = ix; }
            }
            __syncthreads();
        }
        const int fi = redi[0] >> 6, fj = redi[0] & 63;
        const int aa = fi < fj ? fi : fj;
        const int bb = fi < fj ? fj : fi;

        float nr = 0.f;
        if (t < NP) nr = fminf(Dmat[aa * 65 + t], Dmat[bb * 65 + t]); // single linkage
        __syncthreads();
        if (t < NP) {
            Dmat[aa * 65 + t] = nr;
            Dmat[t * 65 + aa] = nr;
            if (cl[t] == bb) cl[t] = aa;
        }
        if (t == 0) act[bb] = 0;
        __syncthreads();
    }

    // ---- stage 6: adjacency mismatch count --------------------------------
    float ec = 0.f;
    for (int idx = t; idx < NP * NP; idx += 256) {
        const int i = idx >> 6, j = idx & 63;
        const int adj  = (cl[i] == cl[j]);
        const int tadj = (tl[i] == tl[j]);
        ec += (adj != tadj) ? 1.f : 0.f;
    }
    red[t] = ec; __syncthreads();
    for (int o = 128; o; o >>= 1) { if (t < o) red[t] += red[t + o]; __syncthreads(); }
    if (t == 0) ws[3 * NB + b] = red[0];
}

// ---- deterministic final reduction over 256 batch partials ----------------
__global__ __launch_bounds__(256) void qscm_finalize_kernel(
    const float* __restrict__ ws, float* __restrict__ out)
{
    __shared__ float s0[256], s1[256], s2[256], s3[256];
    const int t = threadIdx.x;
    s0[t] = ws[t];
    s1[t] = ws[NB + t];
    s2[t] = ws[2 * NB + t];
    s3[t] = ws[3 * NB + t];
    __syncthreads();
    for (int o = 128; o; o >>= 1) {
        if (t < o) {
            s0[t] += s0[t + o];
            s1[t] += s1[t + o];
            s2[t] += s2[t + o];
            s3[t] += s3[t + o];
        }
        __syncthreads();
    }
    if (t == 0) {
        const float totpairs = (float)NB * (float)(NP * NP);
        const float mean_sim = s0[0] / s1[0];
        const float mean_dis = s2[0] / (totpairs - s1[0]);
        const float err_mean = s3[0] / (float)NB;   // mean over batch, sum i,j
        out[0] = err_mean + 0.5f * (mean_sim - mean_dis);
    }
}

extern "C" void kernel_launch(void* const* d_in, const int* in_sizes, int n_in,
                              void* d_out, int out_size, void* d_ws, size_t ws_size,
                              hipStream_t stream) {
    const float* q      = (const float*)d_in[0];  // [256, 768]
    const float* psg    = (const float*)d_in[1];  // [256, 64, 768]
    const int*   labels = (const int*)d_in[2];    // [256, 64] int32
    float* out = (float*)d_out;                   // scalar fp32
    float* ws  = (float*)d_ws;                    // 4*256 floats of partials

    qscm_batch_kernel<<<NB, 256, 0, stream>>>(q, psg, labels, ws);
    qscm_finalize_kernel<<<1, 256, 0, stream>>>(ws, out);
}